// DTW_74474732912689
// MI455X (gfx1250) — compile-verified
//
#include <hip/hip_runtime.h>
#include <hip/hip_bf16.h>
#include <math.h>

typedef __attribute__((ext_vector_type(2))) float v2f;
typedef __attribute__((ext_vector_type(8))) float v8f;

#define BB 64
#define NN 512
#define MM 512
#define DD 256
#define EPSN 1e-8f
#define BIGF 3.0e38f

// ---------------------------------------------------------------------------
// Kernel 1: inverse row norms for x and y.  One wave (32 lanes) per row of
// 256 floats; shuffle tree reduction.  invx[b*N+n] = 1/max(||x||, EPS).
// ---------------------------------------------------------------------------
__global__ __launch_bounds__(256) void norms_kernel(
    const float* __restrict__ x, const float* __restrict__ y,
    float* __restrict__ invx, float* __restrict__ invy) {
  const int wave = (int)((blockIdx.x * blockDim.x + threadIdx.x) >> 5);
  const int lane = (int)(threadIdx.x & 31);
  const int NXROWS = BB * NN;
  const float* src;
  float* dst;
  int row;
  if (wave < NXROWS) { src = x; dst = invx; row = wave; }
  else               { src = y; dst = invy; row = wave - NXROWS; }
  const float* p = src + (size_t)row * DD;
  float s = 0.0f;
#pragma unroll
  for (int k = 0; k < 8; ++k) {
    float v = p[lane + 32 * k];
    s += v * v;
  }
#pragma unroll
  for (int off = 16; off; off >>= 1) s += __shfl_xor(s, off, 32);
  if (lane == 0) {
    float n = fmaxf(sqrtf(s), EPSN);
    dst[row] = 1.0f / n;
  }
}

// ---------------------------------------------------------------------------
// Kernel 2: cost GEMM with V_WMMA_F32_16X16X4_F32.
// Each wave computes a 16(x-rows) x 64(y-rows) strip of cost[b].
//   A (16x4, MxK) layout: lane = khi*16 + m, vgpr v holds K = 2*khi + v
//   B (4x16, KxN) layout: lane = khi*16 + n, vgpr v holds K = 2*khi + v
// -> per lane: one contiguous float2 from x, four contiguous float2 from y.
// Epilogue: cost = 1 - dot * invx[n] * invy[m].
// C/D layout: vgpr v at lane -> row = v + 8*(lane>=16), col = lane&15.
// ---------------------------------------------------------------------------
__global__ __launch_bounds__(256) void cost_gemm_kernel(
    const float* __restrict__ x, const float* __restrict__ y,
    const float* __restrict__ invx, const float* __restrict__ invy,
    float* __restrict__ cost) {
  const int gw   = (int)((blockIdx.x * blockDim.x + threadIdx.x) >> 5);
  const int lane = (int)(threadIdx.x & 31);
  const int b  = gw >> 8;          // 256 wave-jobs per batch
  const int r  = gw & 255;
  const int n0 = (r >> 3) * 16;    // 32 n-tiles
  const int m0 = (r & 7) * 64;     // 8 m-strips
  const int khi = lane >> 4;       // 0/1
  const int lm  = lane & 15;

  const float* xp = x + (size_t)b * NN * DD + (size_t)(n0 + lm) * DD + 2 * khi;
  const float* yb = y + (size_t)b * MM * DD + 2 * khi;

  v8f acc[4];
#pragma unroll
  for (int t = 0; t < 4; ++t) acc[t] = (v8f){0,0,0,0,0,0,0,0};

  for (int k0 = 0; k0 < DD; k0 += 4) {
    v2f afrag = *(const v2f*)(xp + k0);
    v2f b0 = *(const v2f*)(yb + (size_t)(m0 +  0 + lm) * DD + k0);
    v2f b1 = *(const v2f*)(yb + (size_t)(m0 + 16 + lm) * DD + k0);
    v2f b2 = *(const v2f*)(yb + (size_t)(m0 + 32 + lm) * DD + k0);
    v2f b3 = *(const v2f*)(yb + (size_t)(m0 + 48 + lm) * DD + k0);
    acc[0] = __builtin_amdgcn_wmma_f32_16x16x4_f32(false, afrag, false, b0,
                                                   (short)0, acc[0], false, false);
    acc[1] = __builtin_amdgcn_wmma_f32_16x16x4_f32(false, afrag, false, b1,
                                                   (short)0, acc[1], false, false);
    acc[2] = __builtin_amdgcn_wmma_f32_16x16x4_f32(false, afrag, false, b2,
                                                   (short)0, acc[2], false, false);
    acc[3] = __builtin_amdgcn_wmma_f32_16x16x4_f32(false, afrag, false, b3,
                                                   (short)0, acc[3], false, false);
  }

  // epilogue: scale by inverse norms, write cost
  const float* ivx = invx + (size_t)b * NN;
  const float* ivy = invy + (size_t)b * MM;
  float* cb = cost + (size_t)b * NN * MM;
  float ivxl[8];
#pragma unroll
  for (int v = 0; v < 8; ++v) ivxl[v] = ivx[n0 + v + 8 * khi];
#pragma unroll
  for (int t = 0; t < 4; ++t) {
    const int m = m0 + t * 16 + lm;
    const float ivyl = ivy[m];
#pragma unroll
    for (int v = 0; v < 8; ++v) {
      const int n = n0 + v + 8 * khi;
      cb[(size_t)n * MM + m] = 1.0f - acc[t][v] * ivxl[v] * ivyl;
    }
  }
}

// ---------------------------------------------------------------------------
// Kernel 3: DTW cumulative table, anti-diagonal wavefront.
// One 512-thread block per batch; three rotating diagonals live in LDS.
// tc[i,j] = cost[i,j] + min(tc[i-1,j-1], tc[i-1,j], tc[i,j-1])
// ---------------------------------------------------------------------------
__global__ __launch_bounds__(512) void dtw_kernel(
    const float* __restrict__ cost, float* __restrict__ tc) {
  const int b = (int)blockIdx.x;
  const int j = (int)threadIdx.x;   // column index 0..511
  const float* cb = cost + (size_t)b * NN * MM;
  float* tb = tc + (size_t)b * NN * MM;

  __shared__ float dbuf[3][MM];
  float* dcur   = dbuf[0];
  float* dprev  = dbuf[1];
  float* dprev2 = dbuf[2];

  for (int d = 0; d < NN + MM - 1; ++d) {
    const int i = d - j;
    if (i >= 0 && i < NN) {
      const float c = cb[(size_t)i * MM + j];
      float m = BIGF;
      if (i > 0 && j > 0) m = fminf(m, dprev2[j - 1]);  // diag
      if (i > 0)          m = fminf(m, dprev[j]);       // up
      if (j > 0)          m = fminf(m, dprev[j - 1]);   // left
      if (i == 0 && j == 0) m = 0.0f;
      const float v = c + m;
      dcur[j] = v;
      tb[(size_t)i * MM + j] = v;
    }
    __syncthreads();
    float* t2 = dprev2;
    dprev2 = dprev;
    dprev  = dcur;
    dcur   = t2;
  }
}

// ---------------------------------------------------------------------------
// Kernel 4: column sums (neg), backtrack (pos columns), two logsumexps.
// One 512-thread block per batch.  Tie-break matches reference:
// diag if a<=b && a<=c; else up if b<=c; else left.  p[0,0] set (not added)
// at the end, so cost[0][0] is only counted once.
// ---------------------------------------------------------------------------
__global__ __launch_bounds__(512) void finish_kernel(
    const float* __restrict__ cost, const float* __restrict__ tc,
    float* __restrict__ out) {
  const int b = (int)blockIdx.x;
  const int t = (int)threadIdx.x;
  const float* cb = cost + (size_t)b * NN * MM;
  const float* tb = tc + (size_t)b * NN * MM;

  __shared__ float colsum[MM];
  __shared__ float pcol[MM];
  __shared__ float red[MM];

  float s = 0.0f;
  for (int i = 0; i < NN; ++i) s += cb[(size_t)i * MM + t];
  colsum[t] = s;
  pcol[t] = 0.0f;
  __syncthreads();

  // ---- neg = logsumexp(colsum) ----
  red[t] = colsum[t];
  __syncthreads();
  for (int o = 256; o > 0; o >>= 1) {
    if (t < o) red[t] = fmaxf(red[t], red[t + o]);
    __syncthreads();
  }
  float mx = red[0];
  __syncthreads();
  red[t] = expf(colsum[t] - mx);
  __syncthreads();
  for (int o = 256; o > 0; o >>= 1) {
    if (t < o) red[t] += red[t + o];
    __syncthreads();
  }
  const float neg = logf(red[0]) + mx;
  __syncthreads();

  // ---- backtrack (thread 0) ----
  if (t == 0) {
    int i = NN - 1, j = MM - 1;
    pcol[j] += cb[(size_t)i * MM + j];
    while (i > 0 && j > 0) {
      const float a  = tb[(size_t)(i - 1) * MM + (j - 1)];
      const float bu = tb[(size_t)(i - 1) * MM + j];
      const float cl = tb[(size_t)i * MM + (j - 1)];
      const bool dg = (a <= bu) && (a <= cl);
      const bool up = (!dg) && (bu <= cl);
      const int ni = (dg || up) ? i - 1 : i;
      const int nj = (dg || !up) ? j - 1 : j;
      i = ni; j = nj;
      pcol[j] += cb[(size_t)i * MM + j];
    }
    if (!(i == 0 && j == 0)) pcol[0] += cb[0];
  }
  __syncthreads();

  // ---- pos = logsumexp(pcol) ----
  red[t] = pcol[t];
  __syncthreads();
  for (int o = 256; o > 0; o >>= 1) {
    if (t < o) red[t] = fmaxf(red[t], red[t + o]);
    __syncthreads();
  }
  mx = red[0];
  __syncthreads();
  red[t] = expf(pcol[t] - mx);
  __syncthreads();
  for (int o = 256; o > 0; o >>= 1) {
    if (t < o) red[t] += red[t + o];
    __syncthreads();
  }
  if (t == 0) out[b] = (logf(red[0]) + mx) - neg;
}

// ---------------------------------------------------------------------------
extern "C" void kernel_launch(void* const* d_in, const int* in_sizes, int n_in,
                              void* d_out, int out_size, void* d_ws, size_t ws_size,
                              hipStream_t stream) {
  (void)in_sizes; (void)n_in; (void)out_size; (void)ws_size;
  const float* x = (const float*)d_in[0];
  const float* y = (const float*)d_in[1];
  float* out = (float*)d_out;

  float* invx = (float*)d_ws;                        // B*N floats
  float* invy = invx + (size_t)BB * NN;              // B*M floats
  float* cost = invy + (size_t)BB * MM;              // B*N*M floats (64 MB)
  float* tc   = cost + (size_t)BB * NN * MM;         // B*N*M floats (64 MB)

  // 1) inverse norms: one wave per row, 8 waves per block
  {
    const int waves = BB * (NN + MM);                // 65536
    norms_kernel<<<waves / 8, 256, 0, stream>>>(x, y, invx, invy);
  }
  // 2) WMMA cost GEMM: 16384 wave-jobs, 8 waves per block
  {
    const int waves = BB * (NN / 16) * (MM / 64);    // 16384
    cost_gemm_kernel<<<waves / 8, 256, 0, stream>>>(x, y, invx, invy, cost);
  }
  // 3) DTW wavefront: one block per batch
  dtw_kernel<<<BB, 512, 0, stream>>>(cost, tc);
  // 4) reductions + backtrack: one block per batch
  finish_kernel<<<BB, 512, 0, stream>>>(cost, tc, out);
}